// MultiMambaBlock_33157147525913
// MI455X (gfx1250) — compile-verified
//
#include <hip/hip_runtime.h>

typedef float v2f __attribute__((ext_vector_type(2)));
typedef float v8f __attribute__((ext_vector_type(8)));

// Problem constants (B=2, N=2, L=1024, DM=512, D_INNER=1024, D_STATE=16,
// D_CONV=4, DT_RANK=32). BN = B*N = 4, M_total = BN*L = 4096.
#define BN_    4
#define N_GRP  2
#define L_     1024
#define DM_    512
#define DIN_   1024
#define DSTATE 16
#define DTRANK 32
#define MTOT   4096
#define NCHUNK 16      // scan chunks
#define TCH    64      // steps per chunk (NCHUNK*TCH == L_)

// ---------------------------------------------------------------------------
// Generic TN GEMM: C[M,N] = A[M,K] (row-major, lda) * B[N,K]^T (row-major, ldb)
// One wave32 computes a 64x64 tile as a 4x4 grid of 16x16 WMMA tiles:
// per K-step of 4, 8 float2 loads feed 16 V_WMMA_F32_16X16X4_F32 ops
// (A frags reused across 4 N-tiles, B frags across 4 M-tiles).
// epilogue: 0 = none, 1 = +bias[col] then softplus.
// ---------------------------------------------------------------------------
__global__ __launch_bounds__(128)
void wmma_gemm_tn(const float* __restrict__ A, int lda,
                  const float* __restrict__ Bm, int ldb,
                  float* __restrict__ C, int ldc,
                  int M, int N, int K,
                  const float* __restrict__ bias, int epilogue)
{
    const int lane  = threadIdx.x & 31;
    const int wave  = threadIdx.x >> 5;
    const int ntile = N >> 6;                         // 64-wide tiles
    const int tile  = blockIdx.x * 4 + wave;          // wave-uniform
    if (tile >= (M >> 6) * ntile) return;             // uniform guard: EXEC all-1s
    const int tm = tile / ntile;
    const int tn = tile % ntile;

    // A fragment (16x4 f32): lanes 0-15 hold M=lane, K={0,1}; lanes 16-31 hold
    // M=lane-16, K={2,3}.  B fragment (4x16) mirrored: lane holds N=lane&15,
    // same K pair.  Both map to contiguous float2 loads in K.
    const int l15  = lane & 15;
    const int koff = (lane >> 4) << 1;                // 0 or 2
    const size_t astr = (size_t)16 * lda;
    const size_t bstr = (size_t)16 * ldb;
    const float* ap = A  + (size_t)(tm * 64 + l15) * lda + koff;
    const float* bp = Bm + (size_t)(tn * 64 + l15) * ldb + koff;

    v8f acc[4][4] = {};
    for (int k = 0; k < K; k += 4) {
        v2f a[4], b[4];
#pragma unroll
        for (int i = 0; i < 4; ++i) a[i] = *(const v2f*)(ap + i * astr + k);
#pragma unroll
        for (int j = 0; j < 4; ++j) b[j] = *(const v2f*)(bp + j * bstr + k);
#pragma unroll
        for (int i = 0; i < 4; ++i)
#pragma unroll
            for (int j = 0; j < 4; ++j)
                acc[i][j] = __builtin_amdgcn_wmma_f32_16x16x4_f32(
                                false, a[i], false, b[j],
                                (short)0, acc[i][j], false, false);
    }

    // C/D layout: VGPR r -> row +r (lanes 0-15) / +8+r (lanes 16-31)
    const int rb = (lane >> 4) << 3;
#pragma unroll
    for (int i = 0; i < 4; ++i) {
        const int rowb = tm * 64 + i * 16 + rb;
#pragma unroll
        for (int j = 0; j < 4; ++j) {
            const int col = tn * 64 + j * 16 + l15;
            const float bv = (epilogue == 1) ? bias[col] : 0.0f;
#pragma unroll
            for (int r = 0; r < 8; ++r) {
                float v = acc[i][j][r];
                if (epilogue == 1) {
                    v += bv;
                    v = (v > 20.0f) ? v : __logf(1.0f + __expf(v));  // softplus
                }
                C[(size_t)(rowb + r) * ldc + col] = v;
            }
        }
    }
}

// ---------------------------------------------------------------------------
// Depthwise causal conv (k=4) along L + SiLU.
// x_in lives in C1[:, 0:1024] with row stride 2048; channel = n*D_INNER + d.
// ---------------------------------------------------------------------------
__global__ __launch_bounds__(256)
void conv_silu_kernel(const float* __restrict__ C1,
                      const float* __restrict__ cw,
                      const float* __restrict__ cb,
                      float* __restrict__ U, int total)
{
    int idx = blockIdx.x * 256 + threadIdx.x;
    if (idx >= total) return;
    int d  = idx & (DIN_ - 1);
    int l  = (idx >> 10) & (L_ - 1);
    int bn = idx >> 20;
    int n  = bn & (N_GRP - 1);
    int ch = n * DIN_ + d;

    const float* w = cw + (size_t)ch * 4;
    float acc = cb[ch];
    const float* xcol = C1 + ((size_t)bn * L_) * (2 * DIN_) + d;
#pragma unroll
    for (int k = 0; k < 4; ++k) {
        int ll = l - 3 + k;
        float xv = (ll >= 0) ? xcol[(size_t)ll * (2 * DIN_)] : 0.0f;
        acc = fmaf(w[k], xv, acc);
    }
    U[idx] = acc / (1.0f + __expf(-acc));             // silu
}

// ---------------------------------------------------------------------------
// Chunked selective scan, pass 1 (reduce): for each (bn, chunk, d) compute the
// chunk's affine action on the state:  h_end = exp(A_s * S) * h_start + c_s
// where S = sum(dt) over the chunk and c_s = scan result starting from h=0.
// blockIdx encodes (bn, chunk, d-slab) so bn/t are wave-uniform -> the shared
// B_s vectors become scalar (SMEM) loads.
// ---------------------------------------------------------------------------
__global__ __launch_bounds__(128)
void scan_chunk_reduce(const float* __restrict__ DT, const float* __restrict__ U,
                       const float* __restrict__ PR,
                       const float* __restrict__ A_log,
                       float* __restrict__ CH_C, float* __restrict__ CH_S)
{
    const int d  = ((blockIdx.x & 7) << 7) + threadIdx.x;     // 0..1023
    const int ck = (blockIdx.x >> 3) & (NCHUNK - 1);          // uniform
    const int bn = blockIdx.x >> 7;                           // uniform

    float As[DSTATE], h[DSTATE];
#pragma unroll
    for (int s = 0; s < DSTATE; ++s) {
        As[s] = -__expf(A_log[(size_t)d * DSTATE + s]);
        h[s]  = 0.0f;
    }
    float S = 0.0f;
    const int t0 = ck * TCH;
    for (int t = t0; t < t0 + TCH; ++t) {
        const size_t base = (size_t)bn * L_ + t;              // uniform
        const float* pr = PR + base * 64;                     // uniform -> SMEM
        float dtv = DT[base * DIN_ + d];
        float uv  = U [base * DIN_ + d];
        S += dtv;
        float dtu = dtv * uv;
#pragma unroll
        for (int s = 0; s < DSTATE; ++s)
            h[s] = fmaf(__expf(dtv * As[s]), h[s], dtu * pr[DTRANK + s]);
    }
    const size_t cb = (size_t)(bn * NCHUNK + ck) * DSTATE * DIN_ + d;
#pragma unroll
    for (int s = 0; s < DSTATE; ++s) CH_C[cb + (size_t)s * DIN_] = h[s];
    CH_S[(size_t)(bn * NCHUNK + ck) * DIN_ + d] = S;
}

// ---------------------------------------------------------------------------
// Chunked scan, pass 2 (emit): rebuild this chunk's start state from the <=15
// previous chunk summaries (h = exp(A_s*S_j)*h + c_j), then replay the chunk
// emitting y_t = sum_s h_s*C_s, fused with +u*Dp and *silu(res) gating.
// ---------------------------------------------------------------------------
__global__ __launch_bounds__(128)
void scan_chunk_emit(const float* __restrict__ DT, const float* __restrict__ U,
                     const float* __restrict__ PR, const float* __restrict__ C1,
                     const float* __restrict__ A_log, const float* __restrict__ Dp,
                     const float* __restrict__ CH_C, const float* __restrict__ CH_S,
                     float* __restrict__ Y2)
{
    const int d  = ((blockIdx.x & 7) << 7) + threadIdx.x;
    const int ck = (blockIdx.x >> 3) & (NCHUNK - 1);          // uniform
    const int bn = blockIdx.x >> 7;                           // uniform

    float As[DSTATE], h[DSTATE];
#pragma unroll
    for (int s = 0; s < DSTATE; ++s) {
        As[s] = -__expf(A_log[(size_t)d * DSTATE + s]);
        h[s]  = 0.0f;
    }
    // prefix: fold in all previous chunks (uniform trip count)
    for (int j = 0; j < ck; ++j) {
        float Sj = CH_S[(size_t)(bn * NCHUNK + j) * DIN_ + d];
        const size_t cb = (size_t)(bn * NCHUNK + j) * DSTATE * DIN_ + d;
#pragma unroll
        for (int s = 0; s < DSTATE; ++s)
            h[s] = fmaf(__expf(Sj * As[s]), h[s], CH_C[cb + (size_t)s * DIN_]);
    }

    const float dp = Dp[d];
    const int t0 = ck * TCH;
    for (int t = t0; t < t0 + TCH; ++t) {
        const size_t base = (size_t)bn * L_ + t;              // uniform
        const float* pr = PR + base * 64;                     // uniform -> SMEM
        float dtv = DT[base * DIN_ + d];
        float uv  = U [base * DIN_ + d];
        float dtu = dtv * uv;
        float y = 0.0f;
#pragma unroll
        for (int s = 0; s < DSTATE; ++s) {
            h[s] = fmaf(__expf(dtv * As[s]), h[s], dtu * pr[DTRANK + s]);
            y    = fmaf(h[s], pr[DTRANK + DSTATE + s], y);
        }
        float res = C1[base * (2 * DIN_) + DIN_ + d];
        float sr  = res / (1.0f + __expf(-res));              // silu(res)
        Y2[base * DIN_ + d] = (y + uv * dp) * sr;
    }
}

// ---------------------------------------------------------------------------
// Orchestration. Workspace layout (floats):
//   C1   : 4096*2048          (x_and_res)
//   U    : 4096*1024          (silu(conv(x_in)))
//   PR   : 4096*64            (dt_in|B|C proj)
//   DTb  : 4096*1024          (softplus dt)
//   Y2   : 4096*1024          (gated scan output)
//   CH_C : 4*16*16*1024       (chunk affine terms)
//   CH_S : 4*16*1024          (chunk dt sums)
// Total ~89.4 MB.
// ---------------------------------------------------------------------------
extern "C" void kernel_launch(void* const* d_in, const int* in_sizes, int n_in,
                              void* d_out, int out_size, void* d_ws, size_t ws_size,
                              hipStream_t stream)
{
    const float* x      = (const float*)d_in[0];
    const float* W_in   = (const float*)d_in[1];
    const float* conv_w = (const float*)d_in[2];
    const float* conv_b = (const float*)d_in[3];
    const float* W_x    = (const float*)d_in[4];
    const float* W_dt   = (const float*)d_in[5];
    const float* b_dt   = (const float*)d_in[6];
    const float* A_log  = (const float*)d_in[7];
    const float* Dp     = (const float*)d_in[8];
    const float* W_out  = (const float*)d_in[9];
    float* out = (float*)d_out;

    float* C1   = (float*)d_ws;
    float* U    = C1   + (size_t)MTOT * (2 * DIN_);
    float* PR   = U    + (size_t)MTOT * DIN_;
    float* DTb  = PR   + (size_t)MTOT * 64;
    float* Y2   = DTb  + (size_t)MTOT * DIN_;
    float* CH_C = Y2   + (size_t)MTOT * DIN_;
    float* CH_S = CH_C + (size_t)BN_ * NCHUNK * DSTATE * DIN_;

    // 1) x_and_res = x @ W_in^T : (4096x512)@(512x2048)
    int tiles = (MTOT / 64) * ((2 * DIN_) / 64);      // 2048
    wmma_gemm_tn<<<tiles / 4, 128, 0, stream>>>(x, DM_, W_in, DM_,
                                                C1, 2 * DIN_,
                                                MTOT, 2 * DIN_, DM_, nullptr, 0);

    // 2) depthwise conv + silu -> U
    conv_silu_kernel<<<(MTOT * DIN_) / 256, 256, 0, stream>>>(C1, conv_w, conv_b,
                                                              U, MTOT * DIN_);

    // 3) proj = U @ W_x^T : (4096x1024)@(1024x64)
    tiles = (MTOT / 64) * (64 / 64);                  // 64
    wmma_gemm_tn<<<tiles / 4, 128, 0, stream>>>(U, DIN_, W_x, DIN_,
                                                PR, 64,
                                                MTOT, 64, DIN_, nullptr, 0);

    // 4) dt = softplus(PR[:, :32] @ W_dt^T + b_dt) : (4096x32)@(32x1024)
    tiles = (MTOT / 64) * (DIN_ / 64);                // 1024
    wmma_gemm_tn<<<tiles / 4, 128, 0, stream>>>(PR, 64, W_dt, DTRANK,
                                                DTb, DIN_,
                                                MTOT, DIN_, DTRANK, b_dt, 1);

    // 5) chunked selective scan (16x parallelism over plain scan)
    const int scan_blocks = BN_ * NCHUNK * (DIN_ / 128);      // 512
    scan_chunk_reduce<<<scan_blocks, 128, 0, stream>>>(DTb, U, PR, A_log,
                                                       CH_C, CH_S);
    scan_chunk_emit<<<scan_blocks, 128, 0, stream>>>(DTb, U, PR, C1, A_log, Dp,
                                                     CH_C, CH_S, Y2);

    // 6) out = Y2 @ W_out^T : (4096x1024)@(1024x512)
    tiles = (MTOT / 64) * (DM_ / 64);                 // 512
    wmma_gemm_tn<<<tiles / 4, 128, 0, stream>>>(Y2, DIN_, W_out, DIN_,
                                                out, DM_,
                                                MTOT, DM_, DIN_, nullptr, 0);
}